// GCNN_2L_31250182045887
// MI455X (gfx1250) — compile-verified
//
#include <hip/hip_runtime.h>

typedef __attribute__((ext_vector_type(2))) float v2f;
typedef __attribute__((ext_vector_type(8))) float v8f;

#define IN_DIM  128
#define HID_DIM 128
#define OUT_DIM 64

// ---------------------------------------------------------------- utilities
__global__ __launch_bounds__(256) void k_zero4(float* __restrict__ p, int n4) {
  int i = blockIdx.x * blockDim.x + threadIdx.x;
  if (i < n4) ((float4*)p)[i] = make_float4(0.f, 0.f, 0.f, 0.f);
}

__global__ __launch_bounds__(256) void k_degrees(const int* __restrict__ src,
                                                 const int* __restrict__ dst,
                                                 float* __restrict__ deg_out,
                                                 float* __restrict__ deg_in, int E) {
  int i = blockIdx.x * blockDim.x + threadIdx.x;
  if (i < E) {
    atomicAdd(&deg_out[src[i]], 1.0f);
    atomicAdd(&deg_in[dst[i]], 1.0f);
  }
}

__global__ __launch_bounds__(256) void k_rsqrt_clamp(float* __restrict__ p, int n) {
  int i = blockIdx.x * blockDim.x + threadIdx.x;
  if (i < n) p[i] = rsqrtf(fmaxf(p[i], 1.0f));
}

// xs[n][f] = h[f][n] * norm_src[n]   (feature-major -> node-major transpose)
__global__ __launch_bounds__(256) void k_transpose_scale(const float* __restrict__ h,
                                                         const float* __restrict__ norm_src,
                                                         float* __restrict__ xs, int N) {
  __shared__ float tile[32][33];
  const int n0 = blockIdx.x * 32;
  const int f0 = blockIdx.y * 32;
  const int tx = threadIdx.x & 31;
  const int ty = threadIdx.x >> 5;          // 0..7
#pragma unroll
  for (int j = 0; j < 32; j += 8) {
    int n = n0 + tx;
    tile[ty + j][tx] = (n < N) ? h[(size_t)(f0 + ty + j) * N + n] : 0.f;
  }
  __syncthreads();
#pragma unroll
  for (int j = 0; j < 32; j += 8) {
    int n = n0 + ty + j;
    if (n < N) xs[(size_t)n * IN_DIM + f0 + tx] = tile[tx][ty + j] * norm_src[n];
  }
}

// One wave32 per edge: gather 128 floats (float4/lane), atomic-scatter to agg[dst].
__global__ __launch_bounds__(256) void k_aggregate(const int* __restrict__ src,
                                                   const int* __restrict__ dst,
                                                   const float* __restrict__ xs,
                                                   float* __restrict__ agg, int E) {
  int gw = (blockIdx.x * 256 + threadIdx.x) >> 5;
  int lane = threadIdx.x & 31;
  if (gw >= E) return;
  int s = src[gw];
  int d = dst[gw];
  float4 v = ((const float4*)(xs + (size_t)s * IN_DIM))[lane];
  float* ag = agg + (size_t)d * IN_DIM + lane * 4;
  atomicAdd(ag + 0, v.x);
  atomicAdd(ag + 1, v.y);
  atomicAdd(ag + 2, v.z);
  atomicAdd(ag + 3, v.w);
}

// ---------------------------------------------------------------- WMMA GEMMs
// GEMM1: out[n][k] = relu( (agg[n][:] * norm_dst[n]) @ W[:,k] + bias[k] ) * norm_src[n]
// Block: 256 threads = 8 waves; block owns one 16-row tile, wave w owns col tile w.
__global__ __launch_bounds__(256) void k_gemm1_wmma(const float* __restrict__ agg,
                                                    const float* __restrict__ norm_dst,
                                                    const float* __restrict__ norm_src,
                                                    const float* __restrict__ W,
                                                    const float* __restrict__ bias,
                                                    float* __restrict__ out, int N) {
  __shared__ float As[16][IN_DIM + 4];
  const int row0 = blockIdx.x * 16;
  const int tid = threadIdx.x;
  for (int i = tid; i < 16 * (IN_DIM / 4); i += 256) {
    int r = i / (IN_DIM / 4);
    int f4 = i % (IN_DIM / 4);
    int n = row0 + r;
    float4 v = (n < N) ? ((const float4*)(agg + (size_t)n * IN_DIM))[f4]
                       : make_float4(0.f, 0.f, 0.f, 0.f);
    float s = (n < N) ? norm_dst[n] : 0.f;
    As[r][f4 * 4 + 0] = v.x * s;
    As[r][f4 * 4 + 1] = v.y * s;
    As[r][f4 * 4 + 2] = v.z * s;
    As[r][f4 * 4 + 3] = v.w * s;
  }
  __syncthreads();
  const int wave = tid >> 5;
  const int lane = tid & 31;
  const int col0 = wave * 16;
  const int hx = lane & 15;             // M index of A frag / N index of B frag
  const int kb = (lane >> 4) << 1;      // K sub-offset: 0 or 2
  v8f c = {};
#pragma unroll
  for (int k = 0; k < IN_DIM; k += 4) {
    v2f a, b;
    a.x = As[hx][k + kb];
    a.y = As[hx][k + kb + 1];
    b.x = W[(size_t)(k + kb) * HID_DIM + col0 + hx];
    b.y = W[(size_t)(k + kb + 1) * HID_DIM + col0 + hx];
    c = __builtin_amdgcn_wmma_f32_16x16x4_f32(false, a, false, b, (short)0, c, false, false);
  }
  const int col = col0 + hx;
  const int rbase = row0 + ((lane >> 4) << 3);
  const float bk = bias[col];
#pragma unroll
  for (int r = 0; r < 8; ++r) {
    int row = rbase + r;
    if (row < N) {
      float v = c[r] + bk;
      v = v > 0.f ? v : 0.f;
      out[(size_t)row * HID_DIM + col] = v * norm_src[row];
    }
  }
}

// GEMM2: out_T[k][n] = (agg[n][:] * norm_dst[n]) @ W2[:,k] + b2[k]   (transposed store)
// Block: 128 threads = 4 waves; block owns one 16-row tile, wave w owns col tile w (OUT_DIM=64).
__global__ __launch_bounds__(128) void k_gemm2_wmma(const float* __restrict__ agg,
                                                    const float* __restrict__ norm_dst,
                                                    const float* __restrict__ W,
                                                    const float* __restrict__ bias,
                                                    float* __restrict__ out, int N) {
  __shared__ float As[16][HID_DIM + 4];
  const int row0 = blockIdx.x * 16;
  const int tid = threadIdx.x;
  for (int i = tid; i < 16 * (HID_DIM / 4); i += 128) {
    int r = i / (HID_DIM / 4);
    int f4 = i % (HID_DIM / 4);
    int n = row0 + r;
    float4 v = (n < N) ? ((const float4*)(agg + (size_t)n * HID_DIM))[f4]
                       : make_float4(0.f, 0.f, 0.f, 0.f);
    float s = (n < N) ? norm_dst[n] : 0.f;
    As[r][f4 * 4 + 0] = v.x * s;
    As[r][f4 * 4 + 1] = v.y * s;
    As[r][f4 * 4 + 2] = v.z * s;
    As[r][f4 * 4 + 3] = v.w * s;
  }
  __syncthreads();
  const int wave = tid >> 5;
  const int lane = tid & 31;
  const int col0 = wave * 16;
  const int hx = lane & 15;
  const int kb = (lane >> 4) << 1;
  v8f c = {};
#pragma unroll
  for (int k = 0; k < HID_DIM; k += 4) {
    v2f a, b;
    a.x = As[hx][k + kb];
    a.y = As[hx][k + kb + 1];
    b.x = W[(size_t)(k + kb) * OUT_DIM + col0 + hx];
    b.y = W[(size_t)(k + kb + 1) * OUT_DIM + col0 + hx];
    c = __builtin_amdgcn_wmma_f32_16x16x4_f32(false, a, false, b, (short)0, c, false, false);
  }
  const int col = col0 + hx;
  const int rbase = row0 + ((lane >> 4) << 3);
  const float bk = bias[col];
  float* op = out + (size_t)col * N + rbase;
  if (rbase + 8 <= N) {
    ((float4*)op)[0] = make_float4(c[0] + bk, c[1] + bk, c[2] + bk, c[3] + bk);
    ((float4*)op)[1] = make_float4(c[4] + bk, c[5] + bk, c[6] + bk, c[7] + bk);
  } else {
#pragma unroll
    for (int r = 0; r < 8; ++r)
      if (rbase + r < N) op[r] = c[r] + bk;
  }
}

// ---------------------------------------------------------------- launcher
extern "C" void kernel_launch(void* const* d_in, const int* in_sizes, int n_in,
                              void* d_out, int out_size, void* d_ws, size_t ws_size,
                              hipStream_t stream) {
  const float* h   = (const float*)d_in[0];
  const int*   src = (const int*)d_in[1];
  const int*   dst = (const int*)d_in[2];
  const float* W1  = (const float*)d_in[3];
  const float* b1  = (const float*)d_in[4];
  const float* W2  = (const float*)d_in[5];
  const float* b2  = (const float*)d_in[6];
  float* out = (float*)d_out;

  const int N = in_sizes[0] / IN_DIM;   // 50000
  const int E = in_sizes[1];            // 640000

  // workspace layout (floats): [norm_src | norm_dst | bufA (N*128) | bufB (N*128)]
  float* norm_src = (float*)d_ws;
  float* norm_dst = norm_src + N;
  float* bufA = norm_dst + N;                       // xs1, then xs2
  float* bufB = bufA + (size_t)N * IN_DIM;          // agg1, then agg2

  const int nrm4 = (2 * N + 3) / 4;
  const int agg4 = (N * IN_DIM) / 4;

  // degrees -> norms
  k_zero4<<<(nrm4 + 255) / 256, 256, 0, stream>>>(norm_src, nrm4);
  k_degrees<<<(E + 255) / 256, 256, 0, stream>>>(src, dst, norm_src, norm_dst, E);
  k_rsqrt_clamp<<<(2 * N + 255) / 256, 256, 0, stream>>>(norm_src, 2 * N);

  // xs1 = h^T * norm_src
  dim3 tgrid((N + 31) / 32, IN_DIM / 32);
  k_transpose_scale<<<tgrid, 256, 0, stream>>>(h, norm_src, bufA, N);

  // layer 1: aggregate + WMMA GEMM (+b1, relu, * norm_src folded for layer 2)
  k_zero4<<<(agg4 + 255) / 256, 256, 0, stream>>>(bufB, agg4);
  k_aggregate<<<(E * 32 + 255) / 256, 256, 0, stream>>>(src, dst, bufA, bufB, E);
  k_gemm1_wmma<<<(N + 15) / 16, 256, 0, stream>>>(bufB, norm_dst, norm_src, W1, b1, bufA, N);

  // layer 2: aggregate + WMMA GEMM (+b2), transposed store to d_out [64, N]
  k_zero4<<<(agg4 + 255) / 256, 256, 0, stream>>>(bufB, agg4);
  k_aggregate<<<(E * 32 + 255) / 256, 256, 0, stream>>>(src, dst, bufA, bufB, E);
  k_gemm2_wmma<<<(N + 15) / 16, 128, 0, stream>>>(bufB, norm_dst, W2, b2, out, N);
}